// ECCConv_57629871177995
// MI455X (gfx1250) — compile-verified
//
#include <hip/hip_runtime.h>

// CDNA5 wave32 WMMA fragment types for V_WMMA_F32_16X16X4_F32:
// A (16x4 f32) / B (4x16 f32) = 2 VGPRs/lane, C/D (16x16 f32) = 8 VGPRs/lane.
typedef float v2f __attribute__((ext_vector_type(2)));
typedef float v8f __attribute__((ext_vector_type(8)));

#define N_NODES 50000
#define N_EDGES 500000
#define F_IN    16
#define F_OUT   16
#define E_FEAT  8
#define HIDDEN  32
#define N_TILES (N_EDGES / 16)      // 31250 (exact)
#define N_PAIRS (N_TILES / 2)       // 15625 (exact) -- 2 tiles per wave
#define WAVES_PER_BLOCK 4
#define THREADS (WAVES_PER_BLOCK * 32)   // 128
#define W2R_ROWS 528                 // 512 (W2 as [512,16]) + 16 (b2 as [16,16])
#define HROW 34                      // padded h row stride: 68-dword b64 stride,
                                     // halves hit disjoint bank sets (no conflicts)

__device__ __forceinline__ float relu1(float v) {
  return __builtin_amdgcn_fmed3f(v, 0.0f, 3.0e38f);   // single v_med3_num_f32
}

// out[n,:] = x[n,:] @ root_kernel + bias   (initializes d_out)
__global__ __launch_bounds__(256) void ecc_root_kernel(
    const float* __restrict__ x, const float* __restrict__ rk,
    const float* __restrict__ bias, float* __restrict__ out) {
  int n = blockIdx.x * 256 + threadIdx.x;
  if (n >= N_NODES) return;
  float xr[F_IN];
#pragma unroll
  for (int f = 0; f < F_IN; ++f) xr[f] = x[n * F_IN + f];
#pragma unroll
  for (int c = 0; c < F_OUT; ++c) {
    float acc = bias[c];
#pragma unroll
    for (int f = 0; f < F_IN; ++f) acc = fmaf(xr[f], rk[f * F_OUT + c], acc);
    out[n * F_OUT + c] = acc;
  }
}

// Fused edge pipeline, 2 edge-tiles (32 edges) per wave:
//   h = relu(e@W1+b1);  msg = (h (x) x_j) @ reshape(W2,[512,16]) + x_j @ reshape(b2,[16,16])
//   atomic scatter-add to out. Two independent WMMA accumulator chains share each B fragment.
__global__ __launch_bounds__(THREADS) void ecc_edge_kernel(
    const float* __restrict__ x, const float* __restrict__ e,
    const int* __restrict__ idx_i, const int* __restrict__ idx_j,
    const float* __restrict__ W1, const float* __restrict__ b1,
    const float* __restrict__ W2, const float* __restrict__ b2,
    float* __restrict__ out) {
  __shared__ float w2r[W2R_ROWS * 16];                  // 33792 B, shared by block
  // h values for the two tile groups interleaved as adjacent dwords -> b64 LDS ops
  __shared__ float hbuf[WAVES_PER_BLOCK][16 * HROW * 2];

  const int tid = threadIdx.x;

  // ---- Stage W2 ([32,256] row-major == [512,16] row-major) + b2 (rows 512..527)
  // via CDNA5 async global->LDS DMA (ASYNCcnt-tracked, no VGPR round-trip).
  {
    const unsigned lds_base = (unsigned)(uintptr_t)(&w2r[0]);
#pragma unroll
    for (int it = 0; it < 16; ++it) {              // 16*128 = 2048 x B128 = 8192 dwords
      const unsigned ldso = lds_base + (unsigned)((it * THREADS + tid) * 16);
      const float* g = W2 + (size_t)(it * THREADS + tid) * 4;
      asm volatile("global_load_async_to_lds_b128 %0, %1, off"
                   :: "v"(ldso), "v"(g) : "memory");
    }
#pragma unroll
    for (int it = 0; it < 2; ++it) {               // 2*128 = 256 dwords of b2
      const unsigned ldso = lds_base + (unsigned)(512 * 16 * 4 + (it * THREADS + tid) * 4);
      const float* g = b2 + (size_t)(it * THREADS + tid);
      asm volatile("global_load_async_to_lds_b32 %0, %1, off"
                   :: "v"(ldso), "v"(g) : "memory");
    }
    asm volatile("s_wait_asynccnt 0x0" ::: "memory");
  }
  __syncthreads();

  const int wave = tid >> 5;
  const int lane = tid & 31;
  const int m    = lane & 15;      // matrix row (edge-in-tile) / C column
  const int half = lane >> 4;      // 0: K-offsets {0,1}; 1: K-offsets {2,3}
  const int ko   = half * 2;

  int pair = blockIdx.x * WAVES_PER_BLOCK + wave;
  const bool valid = (pair < N_PAIRS);
  if (!valid) pair = 0;            // clamp so WMMA runs with full EXEC; skip writes later

  int tileg[2];
  tileg[0] = pair * 2;
  tileg[1] = pair * 2 + 1;

  // ---- per-lane operand loads for both tile groups
  float2 ep[2][4];                 // e rows (8 f32 each)
  float4 xr[2][4];                 // gathered x_j rows (16 f32 each)
#pragma unroll
  for (int g = 0; g < 2; ++g) {
    const int edge = tileg[g] * 16 + m;
    const float2* e2 = reinterpret_cast<const float2*>(e + (size_t)edge * E_FEAT);
#pragma unroll
    for (int q = 0; q < 4; ++q) ep[g][q] = e2[q];
    const int j = idx_j[edge];
    const float4* xs = reinterpret_cast<const float4*>(x + (size_t)j * F_IN);
#pragma unroll
    for (int q = 0; q < 4; ++q) xr[g][q] = xs[q];
  }

  // ---- h = relu(e_tile[16x8] @ W1[8x32] + b1) for both groups (K=8, 2 steps, 2 col-halves)
#pragma unroll
  for (int nh = 0; nh < 2; ++nh) {
    const int n = nh * 16 + m;
    const float bb = b1[n];
    v2f w1b[2];
#pragma unroll
    for (int kb = 0; kb < 2; ++kb) {
      w1b[kb][0] = W1[(4 * kb + ko) * HIDDEN + n];          // B[4kb+ko,   n]
      w1b[kb][1] = W1[(4 * kb + ko + 1) * HIDDEN + n];      // B[4kb+ko+1, n]
    }
    v8f hcs[2];
#pragma unroll
    for (int g = 0; g < 2; ++g) {
      v8f hc = {bb, bb, bb, bb, bb, bb, bb, bb};            // seed C with bias
#pragma unroll
      for (int kb = 0; kb < 2; ++kb) {
        v2f a;
        a[0] = half ? ep[g][2 * kb + 1].x : ep[g][2 * kb].x;  // A[m, 4kb+ko]
        a[1] = half ? ep[g][2 * kb + 1].y : ep[g][2 * kb].y;  // A[m, 4kb+ko+1]
        hc = __builtin_amdgcn_wmma_f32_16x16x4_f32(false, a, false, w1b[kb], (short)0,
                                                   hc, false, false);
      }
      hcs[g] = hc;
    }
    // relu + transpose via LDS, both groups packed as one b64 store per row element
#pragma unroll
    for (int v = 0; v < 8; ++v) {
      float2 pr;
      pr.x = relu1(hcs[0][v]);
      pr.y = relu1(hcs[1][v]);
      *reinterpret_cast<float2*>(
          &hbuf[wave][((v + 8 * half) * HROW + nh * 16 + m) * 2]) = pr;
    }
  }
  __syncthreads();   // uniform; orders per-wave LDS h stores -> row reads

  // ---- msg = z @ w2r with z[m, h*16+f] = h[m,h]*x_j[m,f]; K = 512 (+16 bias rows).
  // Two independent accumulator chains share each B fragment (1 DS pair per 2 WMMAs);
  // both groups' h scalars arrive with a single ds_load_b64 per hidden index.
  v8f mc0 = {0.f, 0.f, 0.f, 0.f, 0.f, 0.f, 0.f, 0.f};
  v8f mc1 = {0.f, 0.f, 0.f, 0.f, 0.f, 0.f, 0.f, 0.f};
  const float2* hrow =
      reinterpret_cast<const float2*>(&hbuf[wave][(m * HROW) * 2]);
#pragma unroll
  for (int kb4 = 0; kb4 < 32; ++kb4) {       // hidden index h = kb4 (FULL unroll)
    const float2 hv = hrow[kb4];             // {h_g0[m,kb4], h_g1[m,kb4]}
#pragma unroll
    for (int inner = 0; inner < 4; ++inner) { // f block = inner*4 .. +3
      const int kb = kb4 * 4 + inner;
      const int krow = kb * 4 + ko;
      v2f b;
      b[0] = w2r[krow * 16 + m];
      b[1] = w2r[(krow + 1) * 16 + m];
      const float4 x0 = xr[0][inner];
      const float4 x1 = xr[1][inner];
      v2f a0, a1;
      a0[0] = hv.x * (half ? x0.z : x0.x);
      a0[1] = hv.x * (half ? x0.w : x0.y);
      a1[0] = hv.y * (half ? x1.z : x1.x);
      a1[1] = hv.y * (half ? x1.w : x1.y);
      mc0 = __builtin_amdgcn_wmma_f32_16x16x4_f32(false, a0, false, b, (short)0,
                                                  mc0, false, false);
      mc1 = __builtin_amdgcn_wmma_f32_16x16x4_f32(false, a1, false, b, (short)0,
                                                  mc1, false, false);
    }
  }
  // b2 tail: z_ext[:,512+f] = x_j[:,f]; w2r rows 512..527 hold reshape(b2,[16,16])
#pragma unroll
  for (int inner = 0; inner < 4; ++inner) {
    const int krow = 512 + inner * 4 + ko;
    v2f b;
    b[0] = w2r[krow * 16 + m];
    b[1] = w2r[(krow + 1) * 16 + m];
    const float4 x0 = xr[0][inner];
    const float4 x1 = xr[1][inner];
    v2f a0, a1;
    a0[0] = half ? x0.z : x0.x;
    a0[1] = half ? x0.w : x0.y;
    a1[0] = half ? x1.z : x1.x;
    a1[1] = half ? x1.w : x1.y;
    mc0 = __builtin_amdgcn_wmma_f32_16x16x4_f32(false, a0, false, b, (short)0,
                                                mc0, false, false);
    mc1 = __builtin_amdgcn_wmma_f32_16x16x4_f32(false, a1, false, b, (short)0,
                                                mc1, false, false);
  }

  // ---- scatter-add: lane holds msg_g[v + 8*half, m] in mcg[v]
  if (valid) {
#pragma unroll
    for (int v = 0; v < 8; ++v) {
      const int mm = v + 8 * half;
      const int t0 = idx_i[tileg[0] * 16 + mm];
      atomicAdd(&out[(size_t)t0 * F_OUT + m], mc0[v]);
      const int t1 = idx_i[tileg[1] * 16 + mm];
      atomicAdd(&out[(size_t)t1 * F_OUT + m], mc1[v]);
    }
  }
}

extern "C" void kernel_launch(void* const* d_in, const int* in_sizes, int n_in,
                              void* d_out, int out_size, void* d_ws, size_t ws_size,
                              hipStream_t stream) {
  const float* x     = (const float*)d_in[0];
  const float* e     = (const float*)d_in[1];
  const int*   idx_i = (const int*)d_in[2];
  const int*   idx_j = (const int*)d_in[3];
  const float* W1    = (const float*)d_in[4];
  const float* b1    = (const float*)d_in[5];
  const float* W2    = (const float*)d_in[6];
  const float* b2    = (const float*)d_in[7];
  const float* rk    = (const float*)d_in[8];
  const float* bias  = (const float*)d_in[9];
  float* out = (float*)d_out;

  // 1) initialize out with root term (also serves as zero-init for scatter)
  ecc_root_kernel<<<(N_NODES + 255) / 256, 256, 0, stream>>>(x, rk, bias, out);
  // 2) fused edge-conditioned conv + scatter-add (2 tiles per wave)
  const int blocks = (N_PAIRS + WAVES_PER_BLOCK - 1) / WAVES_PER_BLOCK;
  ecc_edge_kernel<<<blocks, THREADS, 0, stream>>>(x, e, idx_i, idx_j, W1, b1, W2, b2, out);
}